// PINN_28664611734313
// MI455X (gfx1250) — compile-verified
//
#include <hip/hip_runtime.h>

typedef __attribute__((ext_vector_type(2))) float v2f;
typedef __attribute__((ext_vector_type(8))) float v8f;

#define N_INT 16384
#define N_BND 4096
#define N_TOT 20480
#define DIMX 3
#define NRES 8
#define NFC 3
#define D 128
#define TPTS 16
#define NTILES (N_TOT / TPTS) /* 1280 */

// Forward jet (u, du/dx, d2u/dx2) through one dimension-net for 16 points.
// GEMMs done with V_WMMA_F32_16X16X4_F32; activations double-buffered in LDS.
__global__ __launch_bounds__(256)
void pinn_dim_forward(const float* __restrict__ X,
                      const float* __restrict__ w_first,
                      const float* __restrict__ b_first,
                      const float* __restrict__ W_res,
                      const float* __restrict__ b_res,
                      const float* __restrict__ W_last,
                      const float* __restrict__ b_last,
                      const float* __restrict__ lb,
                      const float* __restrict__ ub,
                      float* __restrict__ valF,
                      float* __restrict__ d2F)
{
    // [buffer][jet][point][feature]  => 2*3*16*128*4 = 48 KB
    __shared__ float H[2][3][TPTS][D];

    const int dim  = blockIdx.x / NTILES;
    const int tile = blockIdx.x % NTILES;
    const int tid  = threadIdx.x;
    const int w    = tid >> 5;      // wave id 0..7  -> column group
    const int lane = tid & 31;
    const int hl   = lane >> 4;     // lane half
    const int pl   = lane & 15;
    const int n0   = w << 4;        // this wave's 16 output columns

    const float lo = lb[dim], hi = ub[dim];
    const float sc = 2.0f / (hi - lo);          // dt/dx
    const float* wf = w_first + dim * D;
    const float* bf = b_first + dim * D;

    // ---- first layer: h = t*wf + bf ; h' = sc*wf ; h'' = 0 ----
    for (int idx = tid; idx < TPTS * D; idx += 256) {
        int p = idx >> 7, n = idx & (D - 1);
        float x = X[(tile * TPTS + p) * DIMX + dim];
        float t = (x - lo) * sc - 1.0f;
        H[0][0][p][n] = t * wf[n] + bf[n];
        H[0][1][p][n] = sc * wf[n];
        H[0][2][p][n] = 0.0f;
    }
    __syncthreads();

    const float* Wd = W_res + (size_t)dim * NRES * NFC * D * D;
    const float* Bd = b_res + (size_t)dim * NRES * NFC * D;

    int ib = 0;
    for (int res = 0; res < NRES; ++res) {
        // keep the residual (block input) slice in registers:
        // element (jet, M=i+8*hl, N=n0+pl) matches the C/D tile layout
        v8f r0, r1, r2;
        #pragma unroll
        for (int i = 0; i < 8; ++i) {
            int p = i + 8 * hl;
            r0[i] = H[ib][0][p][n0 + pl];
            r1[i] = H[ib][1][p][n0 + pl];
            r2[i] = H[ib][2][p][n0 + pl];
        }
        for (int fc = 0; fc < NFC; ++fc) {
            const float* W = Wd + (size_t)(res * NFC + fc) * D * D; // [k][n]
            const float* B = Bd + (size_t)(res * NFC + fc) * D;
            const int ob = ib ^ 1;
            v8f c0 = {}, c1 = {}, c2 = {};
            #pragma unroll 4
            for (int k0 = 0; k0 < D; k0 += 4) {
                const int kb = k0 + 2 * hl;
                // A fragment (16x4 f32): lane=M, contiguous K pair -> b64 LDS load
                v2f a0 = *(const v2f*)&H[ib][0][pl][kb];
                v2f a1 = *(const v2f*)&H[ib][1][pl][kb];
                v2f a2 = *(const v2f*)&H[ib][2][pl][kb];
                // B fragment (4x16 f32): VGPR v -> K=k0+v+2*hl, N=pl
                v2f bb;
                bb.x = W[(size_t)kb * D + n0 + pl];
                bb.y = W[(size_t)(kb + 1) * D + n0 + pl];
                c0 = __builtin_amdgcn_wmma_f32_16x16x4_f32(
                        false, a0, false, bb, (short)0, c0, false, false);
                c1 = __builtin_amdgcn_wmma_f32_16x16x4_f32(
                        false, a1, false, bb, (short)0, c1, false, false);
                c2 = __builtin_amdgcn_wmma_f32_16x16x4_f32(
                        false, a2, false, bb, (short)0, c2, false, false);
            }
            // epilogue: bias, optional residual add, sin-jet, store to LDS
            float bias = B[n0 + pl];   // N fixed per lane
            #pragma unroll
            for (int i = 0; i < 8; ++i) {
                int p = i + 8 * hl;
                float v  = c0[i] + bias;
                float d1 = c1[i];
                float d2 = c2[i];
                if (fc == NFC - 1) { v += r0[i]; d1 += r1[i]; d2 += r2[i]; }
                float s = __sinf(v);
                float c = __cosf(v);
                H[ob][0][p][n0 + pl] = s;
                H[ob][1][p][n0 + pl] = c * d1;
                H[ob][2][p][n0 + pl] = c * d2 - s * d1 * d1;
            }
            __syncthreads();
            ib = ob;
        }
    }

    // ---- last layer: u = h.Wl + bl ; u'' = h''.Wl  (32 reductions of 128) ----
    const float* Wl = W_last + dim * D;
    const float blv = b_last[dim];
    for (int tt = 0; tt < 4; ++tt) {
        int task = w + tt * 8;            // 0..31
        int p    = task & 15;
        int jet  = (task >> 4) ? 2 : 0;   // value or 2nd derivative
        float sum = 0.f;
        for (int n = lane; n < D; n += 32)
            sum += H[ib][jet][p][n] * Wl[n];
        for (int off = 16; off > 0; off >>= 1)
            sum += __shfl_down(sum, off, 32);
        if (lane == 0) {
            int gp = tile * TPTS + p;
            if (jet == 0) valF[dim * N_TOT + gp] = sum + blv;
            else          d2F[dim * N_TOT + gp] = sum;
        }
    }
}

// equation = -(sum_i F2_i * prod_{j!=i} F_j) - rhs ; boundary = prod_i F_i - rhs
__global__ __launch_bounds__(256)
void pinn_combine(const float* __restrict__ valF,
                  const float* __restrict__ d2F,
                  const float* __restrict__ rhs,
                  float* __restrict__ out)
{
    int n = blockIdx.x * blockDim.x + threadIdx.x;
    if (n >= N_TOT) return;
    float f0 = valF[0 * N_TOT + n];
    float f1 = valF[1 * N_TOT + n];
    float f2 = valF[2 * N_TOT + n];
    if (n < N_INT) {
        float g0 = d2F[0 * N_TOT + n];
        float g1 = d2F[1 * N_TOT + n];
        float g2 = d2F[2 * N_TOT + n];
        float lap = g0 * f1 * f2 + g1 * f0 * f2 + g2 * f0 * f1;
        out[n] = -lap - rhs[n];
    } else {
        out[n] = f0 * f1 * f2 - rhs[n];
    }
}

extern "C" void kernel_launch(void* const* d_in, const int* in_sizes, int n_in,
                              void* d_out, int out_size, void* d_ws, size_t ws_size,
                              hipStream_t stream) {
    const float* X       = (const float*)d_in[0];
    const float* rhs     = (const float*)d_in[1];
    const float* w_first = (const float*)d_in[2];
    const float* b_first = (const float*)d_in[3];
    const float* W_res   = (const float*)d_in[4];
    const float* b_res   = (const float*)d_in[5];
    const float* W_last  = (const float*)d_in[6];
    const float* b_last  = (const float*)d_in[7];
    const float* lbp     = (const float*)d_in[8];
    const float* ubp     = (const float*)d_in[9];

    float* valF = (float*)d_ws;               // 3*20480 floats
    float* d2Fp = valF + DIMX * N_TOT;        // 3*20480 floats

    pinn_dim_forward<<<dim3(DIMX * NTILES), dim3(256), 0, stream>>>(
        X, w_first, b_first, W_res, b_res, W_last, b_last, lbp, ubp, valF, d2Fp);

    pinn_combine<<<dim3((N_TOT + 255) / 256), dim3(256), 0, stream>>>(
        valF, d2Fp, rhs, (float*)d_out);
}